// AggrEncoder_936302870862
// MI455X (gfx1250) — compile-verified
//
#include <hip/hip_runtime.h>

typedef __attribute__((ext_vector_type(2))) float v2f;
typedef __attribute__((ext_vector_type(8))) float v8f;

constexpr int kB    = 32;
constexpr int kT    = 4096;
constexpr int kD    = 128;   // feature dim
constexpr int kTu   = 512;   // time buckets
constexpr int kDout = 64;
constexpr int kM    = 8;     // output dim
constexpr int kMPad = 16;    // W3 rows padded to 16 so B-fragment loads are exec-uniform
constexpr int kPad  = 132;   // padded LDS row stride (132 % 64 == 4 -> conflict-free WMMA A reads)

// LDS budget: 512*132*4 = 264KB hidden + 16*128*4 = 8KB W3 + 32B b3  (< 320KB/WGP)
__global__ __launch_bounds__(256) void aggr_encoder_fused(
    const float* __restrict__ x,            // (B, T, 1+D)
    const unsigned char* __restrict__ mask, // (B, T, 1) bool
    const float* __restrict__ tw_uniq,      // (B, Tu, 1)
    const float* __restrict__ bn_gamma, const float* __restrict__ bn_beta,
    const float* __restrict__ bn_mean,  const float* __restrict__ bn_var,
    const float* __restrict__ conv_w,   const float* __restrict__ conv_b,
    const float* __restrict__ fc_w,     const float* __restrict__ fc_b,
    float* __restrict__ out)                // (B, Tu, M)
{
    extern __shared__ float smem[];
    float* hid = smem;                       // kTu * kPad
    float* w3  = smem + kTu * kPad;          // kMPad * kD (scale-folded fc@conv, rows 8..15 zero)
    float* b3  = w3 + kMPad * kD;            // kM         (fully folded bias)

    const int tid  = threadIdx.x;
    const int b    = blockIdx.x;
    const int lane = tid & 31;
    const int wave = tid >> 5;

    // ---- Phase 0a: zero hidden slab (scatter-max base is 0.0f == int 0) + zero-pad W3
    for (int i = tid; i < kTu * kPad; i += 256) hid[i] = 0.0f;
    for (int i = tid; i < kMPad * kD; i += 256) w3[i] = 0.0f;

    // ---- Phase 0b: constant bias part: fc_b[m] + sum_o fc_w[m,o]*conv_b[o]
    if (tid < kM) {
        float s = fc_b[tid];
        for (int o = 0; o < kDout; ++o) s += fc_w[tid * kDout + o] * conv_b[o];
        b3[tid] = s;
    }
    __syncthreads();

    // ---- Phase 0c: fold BN + conv + fc into W3 (8x128 live rows) and finish b3
    for (int p = tid; p < kM * kD; p += 256) {
        const int m = p >> 7;          // p / kD
        const int d = p & (kD - 1);
        float w2 = 0.0f;
        for (int o = 0; o < kDout; ++o) w2 += fc_w[m * kDout + o] * conv_w[o * kD + d];
        const float sc = bn_gamma[d] * rsqrtf(bn_var[d] + 1e-5f);
        const float sh = bn_beta[d] - bn_mean[d] * sc;
        w3[m * kD + d] = w2 * sc;
        atomicAdd(&b3[m], w2 * sh);
    }

    // ---- Phase 1: scatter-max rows into LDS buckets (ds_max_i32; valid since values >= 0)
    const float  u0    = tw_uniq[(size_t)b * kTu];       // tw_uniq[b,0,0]
    const size_t xbase = (size_t)b * kT * (kD + 1);
    for (int t = wave; t < kT; t += 8) {                 // one row per wave
        const float* xr = x + xbase + (size_t)t * (kD + 1);
        __builtin_prefetch(xr + 8 * (kD + 1), 0, 0);     // next row for this wave
        const unsigned char mk = mask[(size_t)b * kT + t];
        if (!mk) continue;                               // masked rows contribute 0 -> no-op
        int u = (int)(xr[0] - u0);
        u = u < 0 ? 0 : (u > kTu - 1 ? kTu - 1 : u);
        int* brow = (int*)&hid[u * kPad];
#pragma unroll
        for (int j = 0; j < 4; ++j) {
            const int d = lane + 32 * j;                 // 128B-contiguous per j across the wave
            const float v = xr[1 + d];
            if (v > 0.0f) atomicMax(&brow[d], __float_as_int(v)); // int order == float order for >=0
        }
    }
    __syncthreads();

    // ---- Phase 2: out[u, m] = hidden[u, :] . W3[m, :] + b3[m]  via V_WMMA_F32_16X16X4_F32
    // A (16x4 f32): lanes 0-15 carry K=k,k+1 ; lanes 16-31 carry K=k+2,k+3 (ISA 7.12.2)
    const int half = lane >> 4;      // 0 or 1
    const int l15  = lane & 15;

    // Preload all 32 B fragments (tile-invariant): B[k][n] = W3[n][k], rows n>=8 are zeros.
    v2f bfrag[kD / 4];
#pragma unroll
    for (int k4 = 0; k4 < kD / 4; ++k4) {
        const int kk = k4 * 4 + half * 2;
        const float* wr = &w3[l15 * kD + kk];
        bfrag[k4].x = wr[0];
        bfrag[k4].y = wr[1];
    }
    const float bias = (l15 < kM) ? b3[l15] : 0.0f;

    for (int tile = wave; tile < kTu / 16; tile += 8) {  // 32 u-tiles of 16 rows
        const int ub = tile * 16;
        const float* hr = &hid[(ub + l15) * kPad + half * 2];
        v8f acc = {};
#pragma unroll
        for (int k4 = 0; k4 < kD / 4; ++k4) {
            v2f a;
            a.x = hr[k4 * 4];
            a.y = hr[k4 * 4 + 1];
            acc = __builtin_amdgcn_wmma_f32_16x16x4_f32(
                false, a, false, bfrag[k4], (short)0, acc, false, false);
        }
        // D layout: VGPR v, lane l -> row = v + 8*(l>=16), col = l & 15
        if (l15 < kM) {
#pragma unroll
            for (int v = 0; v < 8; ++v) {
                const int row = ub + v + half * 8;
                out[((size_t)b * kTu + row) * kM + l15] = acc[v] + bias;
            }
        }
    }
}

extern "C" void kernel_launch(void* const* d_in, const int* in_sizes, int n_in,
                              void* d_out, int out_size, void* d_ws, size_t ws_size,
                              hipStream_t stream) {
    const float*         x       = (const float*)d_in[0];
    const unsigned char* mask    = (const unsigned char*)d_in[1];
    const float*         tw_uniq = (const float*)d_in[2];
    const float*         g       = (const float*)d_in[3];
    const float*         be      = (const float*)d_in[4];
    const float*         mn      = (const float*)d_in[5];
    const float*         vr      = (const float*)d_in[6];
    const float*         cw      = (const float*)d_in[7];
    const float*         cb      = (const float*)d_in[8];
    const float*         fw      = (const float*)d_in[9];
    const float*         fb      = (const float*)d_in[10];
    float*               out     = (float*)d_out;

    const size_t lds = (size_t)(kTu * kPad + kMPad * kD + kM) * sizeof(float);
    hipFuncSetAttribute((const void*)aggr_encoder_fused,
                        hipFuncAttributeMaxDynamicSharedMemorySize, (int)lds);
    aggr_encoder_fused<<<kB, 256, lds, stream>>>(x, mask, tw_uniq, g, be, mn, vr,
                                                 cw, cb, fw, fb, out);
}